// SAGPooling_69595650064840
// MI455X (gfx1250) — compile-verified
//
#include <hip/hip_runtime.h>
#include <hip/hip_bf16.h>
#include <stdint.h>

#define BLK   256
#define CHUNK 512

// ---------------------------------------------------------------- async glue
typedef int v4i __attribute__((vector_size(16)));
#define AS1 __attribute__((address_space(1)))
#define AS3 __attribute__((address_space(3)))

#if defined(__HIP_DEVICE_COMPILE__) && __has_builtin(__builtin_amdgcn_global_load_async_to_lds_b128) && __has_builtin(__builtin_amdgcn_global_store_async_from_lds_b128)
#define HAVE_ASYNC_LDS 1
#else
#define HAVE_ASYNC_LDS 0
#endif

__device__ __forceinline__ void wait_asynccnt0() {
#if defined(__HIP_DEVICE_COMPILE__)
    asm volatile("s_wait_asynccnt 0" ::: "memory");
#endif
}

// ---------------------------------------------------------------- key build
// key = (~monotone(tanh(score))) << 32 | index  -> ascending sort == topk order
__global__ void build_keys_kernel(const float* __restrict__ score,
                                  uint64_t* __restrict__ keys, int N) {
    int i = blockIdx.x * blockDim.x + threadIdx.x;
    if (i >= N) return;
    float t = tanhf(score[i]);
    uint32_t u = __float_as_uint(t);
    uint32_t mono = (u & 0x80000000u) ? ~u : (u | 0x80000000u);  // float -> ascending uint
    uint32_t desc = ~mono;                                       // descending order
    keys[i] = ((uint64_t)desc << 32) | (uint32_t)i;
}

// ---------------------------------------------------------------- radix sort
__global__ void radix_hist_kernel(const uint64_t* __restrict__ keys,
                                  uint32_t* __restrict__ hist,
                                  int N, int shift, int nblk) {
    __shared__ uint32_t h[256];
    int t = threadIdx.x;
    h[t] = 0;
    __syncthreads();
    int base = blockIdx.x * CHUNK;
    int end  = min(base + CHUNK, N);
    for (int i = base + t; i < end; i += BLK)
        atomicAdd(&h[(uint32_t)(keys[i] >> shift) & 255u], 1u);
    __syncthreads();
    hist[t * nblk + (int)blockIdx.x] = h[t];   // bin-major layout (digit, block)
}

__global__ void radix_scan_kernel(uint32_t* __restrict__ hist, int total) {
    __shared__ uint32_t s[BLK];
    int t = threadIdx.x;
    int items = (total + BLK - 1) / BLK;
    int b0 = t * items;
    int b1 = min(total, b0 + items);
    uint32_t sum = 0;
    for (int i = b0; i < b1; ++i) sum += hist[i];
    s[t] = sum;
    __syncthreads();
    for (int o = 1; o < BLK; o <<= 1) {        // inclusive Hillis-Steele scan
        uint32_t v = (t >= o) ? s[t - o] : 0u;
        __syncthreads();
        s[t] += v;
        __syncthreads();
    }
    uint32_t run = (t == 0) ? 0u : s[t - 1];   // exclusive offset for this thread
    for (int i = b0; i < b1; ++i) {
        uint32_t v = hist[i];
        hist[i] = run;
        run += v;
    }
}

__global__ void radix_scatter_kernel(const uint64_t* __restrict__ src,
                                     uint64_t* __restrict__ dst,
                                     const uint32_t* __restrict__ hist,
                                     int N, int shift, int nblk) {
    __shared__ uint64_t klds[CHUNK];
    __shared__ uint32_t dlds[CHUNK];
    int t = threadIdx.x;
    int base = blockIdx.x * CHUNK;
    int cnt  = min(CHUNK, N - base);
    for (int l = t; l < cnt; l += BLK) {
        uint64_t kv = src[base + l];
        klds[l] = kv;
        dlds[l] = (uint32_t)(kv >> shift) & 255u;
    }
    __syncthreads();
    for (int l = t; l < cnt; l += BLK) {
        uint32_t d = dlds[l];
        uint32_t r = 0;                         // stable in-block rank
        for (int j = 0; j < l; ++j) r += (dlds[j] == d) ? 1u : 0u;
        dst[hist[d * nblk + (int)blockIdx.x] + r] = klds[l];
    }
}

// ---------------------------------------------------------------- extraction
__global__ void extract_kernel(const uint64_t* __restrict__ keys,
                               const float* __restrict__ score,
                               const int* __restrict__ node_type,
                               int* __restrict__ cluster,
                               int* __restrict__ perm_i,
                               float* __restrict__ vals_f,
                               float* __restrict__ out_ntp,
                               float* __restrict__ out_perm,
                               float* __restrict__ out_vals,
                               int N, int k) {
    int r = blockIdx.x * blockDim.x + threadIdx.x;
    if (r >= N) return;
    int idx = (int)(uint32_t)(keys[r] & 0xFFFFFFFFu);
    if (r < k) {
        cluster[idx] = r;
        perm_i[r]    = idx;
        float v = tanhf(score[idx]);
        vals_f[r]   = v;
        out_vals[r] = v;
        out_perm[r] = (float)idx;
        out_ntp[r]  = (float)node_type[idx];
    } else {
        cluster[idx] = -1;
    }
}

// ---------------------------------------------------------------- x gather*scale
__global__ void gather_x_kernel(const float4* __restrict__ x4,
                                const int* __restrict__ perm_i,
                                const float* __restrict__ vals_f,
                                float4* __restrict__ out4,
                                int k, int d4) {
    int tid = blockIdx.x * blockDim.x + threadIdx.x;
    if (tid >= k * d4) return;
    int r = tid / d4;
    int c = tid - r * d4;
    float v = vals_f[r];
    float4 a = x4[(size_t)perm_i[r] * d4 + c];
    a.x *= v; a.y *= v; a.z *= v; a.w *= v;
    out4[tid] = a;
}

// ---------------------------------------------------------------- edge relabel
__global__ void edges_kernel(const int* __restrict__ ei,
                             const float* __restrict__ elen,
                             const int* __restrict__ cluster,
                             float* __restrict__ out_row,
                             float* __restrict__ out_col,
                             float* __restrict__ out_len,
                             float* __restrict__ out_mask,
                             const float* __restrict__ ea, int DE,
                             int E) {
    int e = blockIdx.x * blockDim.x + threadIdx.x;
    if (e >= E) return;
    int cr = cluster[ei[e]];
    int cc = cluster[ei[E + e]];
    bool m = (cr >= 0) && (cc >= 0);
    out_row[e]  = m ? (float)cr : 0.0f;
    out_col[e]  = m ? (float)cc : 0.0f;
    out_len[e]  = m ? elen[e] : 0.0f;
    out_mask[e] = m ? 1.0f : 0.0f;
    __builtin_prefetch(ea + (size_t)e * DE, 0, 1);   // warm L2 for the attr pass
}

// ---------------------------------------------------------------- edge_attr masked copy
// Async DMA path: global -> LDS (per-lane 16B slot), then LDS -> global where
// masked-out lanes source a zeroed LDS slot. Emits gfx1250 async-LDS ops.
__global__ void attr_async_kernel(const float* __restrict__ ea,
                                  const float* __restrict__ mask,
                                  float* __restrict__ out_ea,
                                  int total4, int de4) {
    __shared__ float4 tile[BLK + 1];
    int t = threadIdx.x;
    int g = blockIdx.x * BLK + t;
    if (t == 0) tile[BLK] = make_float4(0.0f, 0.0f, 0.0f, 0.0f);  // zero slot
    bool active = g < total4;
    int e = active ? (g / de4) : 0;
    bool m = active && (mask[e] != 0.0f);
    const float4* gsrc = (const float4*)ea + (active ? g : 0);
    float4* gdst = (float4*)out_ea + (active ? g : 0);
#if HAVE_ASYNC_LDS
    if (active) {
        __builtin_amdgcn_global_load_async_to_lds_b128(
            (AS1 v4i*)(void*)gsrc, (AS3 v4i*)(void*)&tile[t], 0, 0);
    }
    wait_asynccnt0();      // our lane's slot is filled
    __syncthreads();       // zero slot (ds_store by t0) visible block-wide
    if (active) {
        float4* lsrc = m ? &tile[t] : &tile[BLK];
        __builtin_amdgcn_global_store_async_from_lds_b128(
            (AS1 v4i*)(void*)gdst, (AS3 v4i*)(void*)lsrc, 0, 0);
    }
    wait_asynccnt0();
#else
    __syncthreads();
    if (active) {
        float4 v = m ? *gsrc : tile[BLK];
        *gdst = v;
    }
#endif
}

// ---------------------------------------------------------------- launcher
extern "C" void kernel_launch(void* const* d_in, const int* in_sizes, int n_in,
                              void* d_out, int out_size, void* d_ws, size_t ws_size,
                              hipStream_t stream) {
    const float* x            = (const float*)d_in[0];
    const float* score        = (const float*)d_in[1];
    const int*   edge_index   = (const int*)  d_in[2];
    const float* edge_attr    = (const float*)d_in[3];
    const int*   node_type    = (const int*)  d_in[4];
    const float* edge_lengths = (const float*)d_in[5];

    const int N  = in_sizes[1];
    const int D  = in_sizes[0] / N;
    const int E  = in_sizes[5];
    const int DE = in_sizes[3] / E;
    const int k  = (N + 1) / 2;          // ceil(0.5 * N)

    float* out = (float*)d_out;
    size_t off0 = 0;                                // x_p            [k*D]
    size_t off1 = off0 + (size_t)k * D;             // new_edge_index [2*E]
    size_t off2 = off1 + 2 * (size_t)E;             // edge_attr_p    [E*DE]
    size_t off3 = off2 + (size_t)E * DE;            // node_type_p    [k]
    size_t off4 = off3 + (size_t)k;                 // perm           [k]
    size_t off5 = off4 + (size_t)k;                 // vals           [k]
    size_t off6 = off5 + (size_t)k;                 // edge_lengths_p [E]
    size_t off7 = off6 + (size_t)E;                 // edge_mask      [E]

    const int nblk = (N + CHUNK - 1) / CHUNK;

    char* w = (char*)d_ws;
    uint64_t* keysA  = (uint64_t*)w;  w += (size_t)N * 8;
    uint64_t* keysB  = (uint64_t*)w;  w += (size_t)N * 8;
    uint32_t* hist   = (uint32_t*)w;  w += (size_t)nblk * 256 * 4;
    int*      cluster= (int*)w;       w += (size_t)N * 4;
    int*      perm_i = (int*)w;       w += (size_t)k * 4;
    float*    vals_f = (float*)w;     w += (size_t)k * 4;

    // 1. keys
    build_keys_kernel<<<(N + BLK - 1) / BLK, BLK, 0, stream>>>(score, keysA, N);

    // 2. 8-pass LSD radix sort over 64-bit keys (stable per pass)
    uint64_t* src = keysA;
    uint64_t* dst = keysB;
    for (int p = 0; p < 8; ++p) {
        int shift = p * 8;
        radix_hist_kernel   <<<nblk, BLK, 0, stream>>>(src, hist, N, shift, nblk);
        radix_scan_kernel   <<<1,    BLK, 0, stream>>>(hist, nblk * 256);
        radix_scatter_kernel<<<nblk, BLK, 0, stream>>>(src, dst, hist, N, shift, nblk);
        uint64_t* tmp = src; src = dst; dst = tmp;
    }
    // even pass count -> result back in keysA == src

    // 3. perm / vals / node_type_p / cluster
    extract_kernel<<<(N + BLK - 1) / BLK, BLK, 0, stream>>>(
        src, score, node_type, cluster, perm_i, vals_f,
        out + off3, out + off4, out + off5, N, k);

    // 4. x_p = x[perm] * vals
    const int d4 = D / 4;
    gather_x_kernel<<<((size_t)k * d4 + BLK - 1) / BLK, BLK, 0, stream>>>(
        (const float4*)x, perm_i, vals_f, (float4*)(out + off0), k, d4);

    // 5. edge relabel + mask + lengths (+ L2 prefetch of edge_attr)
    edges_kernel<<<(E + BLK - 1) / BLK, BLK, 0, stream>>>(
        edge_index, edge_lengths, cluster,
        out + off1, out + off1 + E, out + off6, out + off7,
        edge_attr, DE, E);

    // 6. masked edge_attr copy via async global<->LDS DMA
    const int total4 = E * (DE / 4);
    attr_async_kernel<<<(total4 + BLK - 1) / BLK, BLK, 0, stream>>>(
        edge_attr, out + off7, out + off2, total4, DE / 4);
}